// VirtualNodePyg_9053791060065
// MI455X (gfx1250) — compile-verified
//
#include <hip/hip_runtime.h>

typedef __attribute__((ext_vector_type(2))) float v2f;
typedef __attribute__((ext_vector_type(4))) float v4f;
typedef __attribute__((ext_vector_type(8))) float v8f;

#define DIM   128
#define DIM_E 64
#define KFC   192   // DIM + DIM_E

// ---------------- zero scratch (only pool_e needs it; atomically accumulated) ----
__global__ void zero_f32(float* __restrict__ p, int n) {
    int i = blockIdx.x * blockDim.x + threadIdx.x;
    if (i < n) p[i] = 0.0f;
}

// ---------------- per-graph node pooling: batch is sorted, 1 block per graph ----
__global__ void pool_nodes_kernel(const float* __restrict__ h,
                                  const int* __restrict__ batch,
                                  int N, float* __restrict__ pool_n) {
    __shared__ int s_lo, s_hi;
    int g = blockIdx.x;
    if (threadIdx.x == 0) {
        int lo = 0, hi = N;
        while (lo < hi) { int mid = (lo + hi) >> 1; if (batch[mid] < g) lo = mid + 1; else hi = mid; }
        s_lo = lo;
        lo = 0; hi = N;
        while (lo < hi) { int mid = (lo + hi) >> 1; if (batch[mid] < g + 1) lo = mid + 1; else hi = mid; }
        s_hi = lo;
    }
    __syncthreads();
    int d = threadIdx.x;                 // 0..127, coalesced across the row
    float acc = 0.0f;
    for (int r = s_lo; r < s_hi; ++r)    // streaming read of h: non-temporal
        acc += __builtin_nontemporal_load(&h[(size_t)r * DIM + d]);
    pool_n[(size_t)g * DIM + d] = acc;
}

// ---------------- edge pooling via f32 global atomics (edge->graph unsorted) ----
__global__ void pool_edges_kernel(const float* __restrict__ e,
                                  const int* __restrict__ ei_src,   // edge_index row 0
                                  const int* __restrict__ batch,
                                  long long total, float* __restrict__ pool_e) {
    long long tid = (long long)blockIdx.x * blockDim.x + threadIdx.x;
    if (tid >= total) return;
    int edge = (int)(tid >> 6);
    int col  = (int)(tid & 63);
    int g = batch[ei_src[edge]];
    float ev = __builtin_nontemporal_load(&e[(size_t)edge * DIM_E + col]); // streaming
    atomicAdd(&pool_e[(size_t)g * DIM_E + col], ev);
}

// ---------------- A = vn_h + concat(pool_n, pool_e) ----------------------------
__global__ void prep_A_kernel(const float* __restrict__ vn_h,
                              const float* __restrict__ pool_n,
                              const float* __restrict__ pool_e,
                              int total, float* __restrict__ Apre) {
    int tid = blockIdx.x * blockDim.x + threadIdx.x;
    if (tid >= total) return;
    int b = tid / KFC, j = tid % KFC;
    float p = (j < DIM) ? pool_n[(size_t)b * DIM + j]
                        : pool_e[(size_t)b * DIM_E + (j - DIM)];
    Apre[tid] = vn_h[tid] + p;
}

// ---------------- WMMA f32 GEMM: C = relu(A[MxK] @ W[KxN] + bias[N]) -----------
// One wave (32 lanes) per 16x16 output tile, V_WMMA_F32_16X16X4_F32, K stepped by 4.
__global__ void wmma_gemm_bias_relu(const float* __restrict__ A,
                                    const float* __restrict__ W,
                                    const float* __restrict__ bias,
                                    float* __restrict__ C,
                                    int K, int N) {
    int lane = threadIdx.x;          // 32 threads per block = 1 wave, EXEC all ones
    int half = lane >> 4;            // 0: lanes 0-15, 1: lanes 16-31
    int l    = lane & 15;
    int n0 = blockIdx.x * 16;
    int m0 = blockIdx.y * 16;
    int arow = m0 + l;               // A: M striped over lanes 0-15 (dup in 16-31)
    int bcol = n0 + l;               // B: N striped over lanes 0-15 (dup in 16-31)

    v8f c = {};
    for (int k = 0; k < K; k += 4) {
        int kb = k + 2 * half;       // lane-half selects K pair (ISA 16x4 f32 layout)
        v2f a, b;
        a.x = A[(size_t)arow * K + kb];
        a.y = A[(size_t)arow * K + kb + 1];
        b.x = W[(size_t)kb       * N + bcol];
        b.y = W[(size_t)(kb + 1) * N + bcol];
        c = __builtin_amdgcn_wmma_f32_16x16x4_f32(
                /*neg_a=*/false, a, /*neg_b=*/false, b,
                /*c_mod=*/(short)0, c, /*reuse_a=*/false, /*reuse_b=*/false);
    }
    float bv = bias[bcol];
#pragma unroll
    for (int v = 0; v < 8; ++v) {
        int row = m0 + v + 8 * half; // C/D layout: VGPR v -> M = v (+8 for hi half)
        float val = c[v] + bv;
        C[(size_t)row * N + bcol] = val > 0.0f ? val : 0.0f;
    }
}

// ---------------- h_out = h + node_msg[batch]  (v4f-vectorized, NT stream) -----
__global__ void bcast_nodes_kernel(const float* __restrict__ h,
                                   const float* __restrict__ node_msg,
                                   const int* __restrict__ batch,
                                   long long total, float* __restrict__ out) {
    long long tid = (long long)blockIdx.x * blockDim.x + threadIdx.x;
    if (tid >= total) return;                 // total = N * 32 (32 v4f per row)
    int node = (int)(tid >> 5);
    int c4   = (int)(tid & 31);
    int g = batch[node];
    // streaming tensors: non-temporal (protect L2 for the hot msg tables)
    v4f a = __builtin_nontemporal_load(((const v4f*)h) + (size_t)node * 32 + c4);
    v4f m = ((const v4f*)node_msg)[(size_t)g * 32 + c4];      // hot 4MB table: RT
    __builtin_nontemporal_store(a + m, ((v4f*)out) + (size_t)node * 32 + c4);
}

// ---------------- e_out = e + edge_msg[batch[src]]  (v4f-vectorized, NT stream) -
__global__ void bcast_edges_kernel(const float* __restrict__ e,
                                   const float* __restrict__ edge_msg,
                                   const int* __restrict__ ei_src,
                                   const int* __restrict__ batch,
                                   long long total, float* __restrict__ out) {
    long long tid = (long long)blockIdx.x * blockDim.x + threadIdx.x;
    if (tid >= total) return;                 // total = E * 16 (16 v4f per row)
    int edge = (int)(tid >> 4);
    int c4   = (int)(tid & 15);
    int g = batch[ei_src[edge]];
    v4f a = __builtin_nontemporal_load(((const v4f*)e) + (size_t)edge * 16 + c4);
    v4f m = ((const v4f*)edge_msg)[(size_t)g * 16 + c4];      // hot 2MB table: RT
    __builtin_nontemporal_store(a + m, ((v4f*)out) + (size_t)edge * 16 + c4);
}

extern "C" void kernel_launch(void* const* d_in, const int* in_sizes, int n_in,
                              void* d_out, int out_size, void* d_ws, size_t ws_size,
                              hipStream_t stream) {
    const float* h     = (const float*)d_in[0];   // [N,128]
    const float* vn_h  = (const float*)d_in[1];   // [B,192]
    const float* e     = (const float*)d_in[2];   // [E,64]
    const float* W_fc  = (const float*)d_in[3];   // [192,128]
    const float* b_fc  = (const float*)d_in[4];   // [128]
    const float* W_np  = (const float*)d_in[5];   // [128,128]
    const float* b_np  = (const float*)d_in[6];   // [128]
    const float* W_ep  = (const float*)d_in[7];   // [128,64]
    const float* b_ep  = (const float*)d_in[8];   // [64]
    const int*   batch = (const int*)d_in[9];     // [N], sorted
    const int*   ei    = (const int*)d_in[10];    // [2,E]; row 0 = src

    const int N = in_sizes[0] / DIM;
    const int B = in_sizes[1] / KFC;
    const int E = in_sizes[2] / DIM_E;

    // workspace layout (floats)
    float* ws       = (float*)d_ws;
    float* pool_n   = ws;                               // B*128
    float* pool_e   = pool_n + (size_t)B * DIM;         // B*64
    float* Apre     = pool_e + (size_t)B * DIM_E;       // B*192
    float* node_msg = Apre   + (size_t)B * KFC;         // B*128
    float* edge_msg = node_msg + (size_t)B * DIM;       // B*64

    // output layout: h_out [N*128] | vn_new [B*128] | e_out [E*64]
    float* h_out  = (float*)d_out;
    float* vn_new = h_out  + (size_t)N * DIM;
    float* e_out  = vn_new + (size_t)B * DIM;

    // 1) zero edge pool (accumulated atomically)
    zero_f32<<<(B * DIM_E + 255) / 256, 256, 0, stream>>>(pool_e, B * DIM_E);

    // 2) node pooling (sorted segments, no atomics)
    pool_nodes_kernel<<<B, DIM, 0, stream>>>(h, batch, N, pool_n);

    // 3) edge pooling (atomic scatter-add)
    long long etot = (long long)E * DIM_E;
    pool_edges_kernel<<<(unsigned)((etot + 255) / 256), 256, 0, stream>>>(e, ei, batch, etot, pool_e);

    // 4) A = vn_h + pool
    prep_A_kernel<<<(B * KFC + 255) / 256, 256, 0, stream>>>(vn_h, pool_n, pool_e, B * KFC, Apre);

    // 5) vn_new = relu(A @ W_fc + b_fc)        [8192,192]x[192,128]
    wmma_gemm_bias_relu<<<dim3(DIM / 16, B / 16), 32, 0, stream>>>(Apre, W_fc, b_fc, vn_new, KFC, DIM);

    // 6) node_msg = relu(vn_new @ W_np + b_np) [8192,128]x[128,128]
    wmma_gemm_bias_relu<<<dim3(DIM / 16, B / 16), 32, 0, stream>>>(vn_new, W_np, b_np, node_msg, DIM, DIM);

    // 7) edge_msg = relu(vn_new @ W_ep + b_ep) [8192,128]x[128,64]
    wmma_gemm_bias_relu<<<dim3(DIM_E / 16, B / 16), 32, 0, stream>>>(vn_new, W_ep, b_ep, edge_msg, DIM, DIM_E);

    // 8) residual broadcasts (bandwidth-dominant, NT-streamed, v4f coalesced)
    long long ntot = (long long)N * 32;
    bcast_nodes_kernel<<<(unsigned)((ntot + 255) / 256), 256, 0, stream>>>(h, node_msg, batch, ntot, h_out);
    long long btot = (long long)E * 16;
    bcast_edges_kernel<<<(unsigned)((btot + 255) / 256), 256, 0, stream>>>(e, edge_msg, ei, batch, btot, e_out);
}